// MaskBatchNorm2d_62414464746114
// MI455X (gfx1250) — compile-verified
//
#include <hip/hip_runtime.h>
#include <hip/hip_bf16.h>
#include <stdint.h>

typedef __attribute__((ext_vector_type(2))) float v2f;
typedef __attribute__((ext_vector_type(8))) float v8f;

#define EPS 1e-5f

// Problem dims (fixed by the reference)
#define BB     32
#define DD     256
#define HW     4096          // 32*128, contiguous per (b,d)
#define GROUPS 16            // DD / 16 channels per WMMA tile
#define SCHUNK 2048          // HW / 2  -> 2 spatial chunks per b
#define SPLITS 64            // BB * 2 K-splits per channel group
#define TILE_S 512           // staged spatial tile per block iteration
#define TILE_PAD 516         // +4 floats row pad -> conflict-free strided LDS reads
#define NTILES (SCHUNK / TILE_S)

// Workspace layout (in floats)
#define WS_SUM   0                       // [GROUPS][SPLITS][16]
#define WS_SQ    (GROUPS*SPLITS*16)      // [GROUPS][SPLITS][16]
#define WS_SCALE (2*GROUPS*SPLITS*16)    // [256]
#define WS_BIAS  (WS_SCALE + 256)        // [256]
#define WS_CNT   (WS_BIAS + 256)         // [1]

// ---------------- Pass 0: deterministic unmasked count -----------------
__global__ void __launch_bounds__(1024)
count_kernel(const unsigned char* __restrict__ mask, float* __restrict__ ws) {
    __shared__ int red[1024];
    const int t = threadIdx.x;
    const uint4* m4 = (const uint4*)mask;          // 131072 bytes = 8192 uint4
    int acc = 0;
#pragma unroll
    for (int j = 0; j < 8; ++j) {
        uint4 v = m4[j * 1024 + t];
        unsigned int wds[4] = {v.x, v.y, v.z, v.w};
#pragma unroll
        for (int k = 0; k < 4; ++k) {
            unsigned int w = wds[k];
            acc += ((w        & 0xffu) == 0u);
            acc += (((w >> 8) & 0xffu) == 0u);
            acc += (((w >> 16)& 0xffu) == 0u);
            acc += (((w >> 24)       ) == 0u);
        }
    }
    red[t] = acc;
    __syncthreads();
    for (int s = 512; s > 0; s >>= 1) {
        if (t < s) red[t] += red[t + s];
        __syncthreads();
    }
    if (t == 0) ws[WS_CNT] = fmaxf((float)red[0], 1.0f);
}

// ---------------- Pass 1: per-channel sum / sumsq via WMMA -------------
// grid = (GROUPS, BB, 2); block = 256 (8 waves of 32)
// Double-buffered LDS tiles filled by global_load_async_to_lds_b128,
// consumed as fp32 16x16x4 WMMA A-tiles against an all-ones B.
__global__ void __launch_bounds__(256)
stats_kernel(const float* __restrict__ x, const unsigned char* __restrict__ mask,
             float* __restrict__ ws) {
    __shared__ __align__(16) float tile[2][16][TILE_PAD];
    __shared__ __align__(16) float wbuf[2][TILE_S];
    __shared__ float redsum[8][16];
    __shared__ float redsq[8][16];

    const int g   = blockIdx.x;      // channel group: channels [16g, 16g+16)
    const int b   = blockIdx.y;
    const int sc  = blockIdx.z;      // spatial half
    const int tid = threadIdx.x;
    const int lane = tid & 31;
    const int wv   = tid >> 5;       // wave id 0..7
    const int m    = lane & 15;      // WMMA A row: M = lane%16
    const int hi   = lane >> 4;      // lanes 16..31 carry K=2,3

    const float* xbase = x + ((size_t)(b * DD + g * 16)) * HW + sc * SCHUNK;
    const unsigned char* mbase = mask + (size_t)b * HW + sc * SCHUNK;

    v8f csum0 = {}, csum1 = {};
    v8f csq0  = {}, csq1  = {};
    v2f ones; ones.x = 1.0f; ones.y = 1.0f;   // B = all-ones 4x16 -> row sums

    // Issue async global->LDS copies + build the 0/1 weight strip for tile t4.
    auto stage = [&](int t4) {
        const int bufsel = t4 & 1;
        const int s0 = t4 * TILE_S;
#pragma unroll
        for (int j = 0; j < 8; ++j) {
            int idx  = j * 256 + tid;          // 2048 float4 slots
            int row  = idx >> 7;               // 128 float4 per channel row
            int col4 = idx & 127;
            const float* gp = xbase + (size_t)row * HW + s0 + col4 * 4;
            unsigned lds_off = (unsigned)(uintptr_t)&tile[bufsel][row][col4 * 4];
            asm volatile("global_load_async_to_lds_b128 %0, %1, off"
                         :: "v"(lds_off), "v"(gp)
                         : "memory");
        }
        if (tid < TILE_S / 4) {
            unsigned mw = *(const unsigned*)(mbase + s0 + tid * 4);
            float4 w;
            w.x = (mw         & 0xffu) ? 0.0f : 1.0f;
            w.y = ((mw >> 8)  & 0xffu) ? 0.0f : 1.0f;
            w.z = ((mw >> 16) & 0xffu) ? 0.0f : 1.0f;
            w.w = ((mw >> 24)        ) ? 0.0f : 1.0f;
            *(float4*)&wbuf[bufsel][tid * 4] = w;
        }
    };

    stage(0);
    for (int t4 = 0; t4 < NTILES; ++t4) {
        asm volatile("s_wait_asynccnt 0x0" ::: "memory");
        __syncthreads();                       // tile t4 fully resident
        if (t4 + 1 < NTILES) stage(t4 + 1);    // prefetch next tile into other buffer

        const int bufsel = t4 & 1;
        const float* trow = &tile[bufsel][m][wv * 64 + 2 * hi];
        const float* wrow = &wbuf[bufsel][wv * 64 + 2 * hi];
#pragma unroll
        for (int k0 = 0; k0 < 64; k0 += 8) {
            v2f x0 = *(const v2f*)(trow + k0);
            v2f w0 = *(const v2f*)(wrow + k0);
            v2f x1 = *(const v2f*)(trow + k0 + 4);
            v2f w1 = *(const v2f*)(wrow + k0 + 4);
            v2f a0 = x0 * w0;
            v2f a1 = x1 * w1;
            v2f q0 = a0 * a0;
            v2f q1 = a1 * a1;
            csum0 = __builtin_amdgcn_wmma_f32_16x16x4_f32(
                false, a0, false, ones, (short)0, csum0, false, false);
            csq0  = __builtin_amdgcn_wmma_f32_16x16x4_f32(
                false, q0, false, ones, (short)0, csq0,  false, false);
            csum1 = __builtin_amdgcn_wmma_f32_16x16x4_f32(
                false, a1, false, ones, (short)0, csum1, false, false);
            csq1  = __builtin_amdgcn_wmma_f32_16x16x4_f32(
                false, q1, false, ones, (short)0, csq1,  false, false);
        }
        // Buffer reuse hazard is covered by the wait+barrier at the top of the
        // next iteration (stage(t4+2) only issues after that barrier).
    }

    // C column N=0 lives in lanes with lane%16==0: vgpr r -> channel r + 8*hi.
    if (m == 0) {
#pragma unroll
        for (int r = 0; r < 8; ++r) {
            redsum[wv][hi * 8 + r] = csum0[r] + csum1[r];
            redsq [wv][hi * 8 + r] = csq0[r]  + csq1[r];
        }
    }
    __syncthreads();

    if (tid < 16) {
        float s = 0.0f, q = 0.0f;
#pragma unroll
        for (int w = 0; w < 8; ++w) { s += redsum[w][tid]; q += redsq[w][tid]; }
        const int split = b * 2 + sc;
        ws[WS_SUM + (g * SPLITS + split) * 16 + tid] = s;
        ws[WS_SQ  + (g * SPLITS + split) * 16 + tid] = q;
    }
}

// ---------------- Pass 2: finalize per-channel scale/bias --------------
__global__ void __launch_bounds__(256)
finalize_kernel(const float* __restrict__ gamma, const float* __restrict__ beta,
                float* __restrict__ ws) {
    const int d = threadIdx.x;
    const int g = d >> 4, m = d & 15;
    float s = 0.0f, q = 0.0f;
    for (int split = 0; split < SPLITS; ++split) {   // fixed order -> deterministic
        s += ws[WS_SUM + (g * SPLITS + split) * 16 + m];
        q += ws[WS_SQ  + (g * SPLITS + split) * 16 + m];
    }
    const float cnt  = ws[WS_CNT];
    const float mean = s / cnt;
    const float var  = fmaxf(q / cnt - mean * mean, 0.0f);
    const float scl  = gamma[d] * rsqrtf(var + EPS);
    ws[WS_SCALE + d] = scl;
    ws[WS_BIAS  + d] = beta[d] - mean * scl;
}

// ---------------- Pass 3: streaming normalize --------------------------
// 8192 blocks * 256 threads * 4 float4 = 33554432 elements
__global__ void __launch_bounds__(256)
norm_kernel(const float* __restrict__ x, const unsigned char* __restrict__ mask,
            const float* __restrict__ ws, float* __restrict__ out) {
#pragma unroll
    for (int j = 0; j < 4; ++j) {
        size_t q = (size_t)blockIdx.x * 1024 + (size_t)j * 256 + threadIdx.x;
        size_t e = q * 4;
        int s = (int)(e & (HW - 1));
        int d = (int)((e >> 12) & 255);
        int b = (int)(e >> 20);
        float4 xv = *(const float4*)(x + e);
        unsigned int mw = *(const unsigned int*)(mask + (size_t)b * HW + s);
        float scl = ws[WS_SCALE + d];
        float bia = ws[WS_BIAS + d];
        float4 o;
        o.x = (mw        & 0xffu) ? xv.x : fmaf(xv.x, scl, bia);
        o.y = ((mw >> 8) & 0xffu) ? xv.y : fmaf(xv.y, scl, bia);
        o.z = ((mw >> 16)& 0xffu) ? xv.z : fmaf(xv.z, scl, bia);
        o.w = ((mw >> 24)       ) ? xv.w : fmaf(xv.w, scl, bia);
        *(float4*)(out + e) = o;
    }
}

extern "C" void kernel_launch(void* const* d_in, const int* in_sizes, int n_in,
                              void* d_out, int out_size, void* d_ws, size_t ws_size,
                              hipStream_t stream) {
    const float*         x     = (const float*)d_in[0];
    const unsigned char* mask  = (const unsigned char*)d_in[1];  // jax bool = 1 byte
    const float*         gamma = (const float*)d_in[2];
    const float*         beta  = (const float*)d_in[3];
    float* ws  = (float*)d_ws;
    float* out = (float*)d_out;

    count_kernel   <<<1, 1024, 0, stream>>>(mask, ws);
    stats_kernel   <<<dim3(GROUPS, BB, 2), 256, 0, stream>>>(x, mask, ws);
    finalize_kernel<<<1, 256, 0, stream>>>(gamma, beta, ws);
    norm_kernel    <<<8192, 256, 0, stream>>>(x, mask, ws, out);
}